// GNNEncoderLayer_56375740727935
// MI455X (gfx1250) — compile-verified
//
#include <hip/hip_runtime.h>

typedef __attribute__((ext_vector_type(2))) float v2f;
typedef __attribute__((ext_vector_type(8))) float v8f;

#define D 512
#define LDS_STRIDE (D + 4)   // 516: pad so 16 A-fragment lanes hit distinct LDS banks

// ---------------- zero workspace ----------------
__global__ void gnn_zero_kernel(float* __restrict__ p, int n) {
  int i = blockIdx.x * blockDim.x + threadIdx.x;
  if (i < n) p[i] = 0.0f;
}

// ---------------- LayerNorm: one wave32 per node ----------------
__global__ void gnn_ln_kernel(const float* __restrict__ xin,
                              const float* __restrict__ gamma,
                              const float* __restrict__ beta,
                              float* __restrict__ xout, int N) {
  int wave = threadIdx.x >> 5;
  int lane = threadIdx.x & 31;
  int node = blockIdx.x * 8 + wave;
  if (node >= N) return;
  const float* row = xin + (size_t)node * D;
  float v[16];
  float s = 0.0f;
#pragma unroll
  for (int i = 0; i < 16; ++i) { v[i] = row[lane + 32 * i]; s += v[i]; }
#pragma unroll
  for (int m = 16; m >= 1; m >>= 1) s += __shfl_xor(s, m, 32);
  float mu = s * (1.0f / D);
  float q = 0.0f;
#pragma unroll
  for (int i = 0; i < 16; ++i) { float d = v[i] - mu; q += d * d; }
#pragma unroll
  for (int m = 16; m >= 1; m >>= 1) q += __shfl_xor(q, m, 32);
  float rstd = rsqrtf(q * (1.0f / D) + 1e-5f);
  float* o = xout + (size_t)node * D;
#pragma unroll
  for (int i = 0; i < 16; ++i) {
    int c = lane + 32 * i;
    o[c] = (v[i] - mu) * rstd * gamma[c] + beta[c];
  }
}

// ---------------- fused scatter-sum + degree count ----------------
// thread -> (edge, 4-float chunk); 128 chunks per edge, L2-side f32 atomics
__global__ void gnn_scatter_kernel(const float* __restrict__ x,
                                   const int* __restrict__ ei,
                                   float* __restrict__ summed,
                                   float* __restrict__ cnt,
                                   long long E) {
  long long idx = (long long)blockIdx.x * blockDim.x + threadIdx.x;
  if (idx >= E * 128) return;
  int c = (int)(idx & 127);
  long long e = idx >> 7;
  size_t src = (size_t)ei[e];
  size_t dst = (size_t)ei[E + e];
  const float4 m = *(const float4*)(x + src * D + c * 4);
  float* sp = summed + dst * D + c * 4;
  unsafeAtomicAdd(sp + 0, m.x);
  unsafeAtomicAdd(sp + 1, m.y);
  unsafeAtomicAdd(sp + 2, m.z);
  unsafeAtomicAdd(sp + 3, m.w);
  if (c == 0) unsafeAtomicAdd(cnt + dst, 1.0f);
}

// ---------------- WMMA GEMM: out = relu(agg@wl^T + bl + x@wr^T) + resid ----
__global__ void __launch_bounds__(256)
gnn_gemm_kernel(const float* __restrict__ xln, const float* __restrict__ summed,
                const float* __restrict__ cnt,
                const float* __restrict__ wl, const float* __restrict__ wr,
                const float* __restrict__ bl, const float* __restrict__ resid,
                float* __restrict__ out) {
  __shared__ float sA[16 * LDS_STRIDE];  // agg panel (mean-scaled)
  __shared__ float sX[16 * LDS_STRIDE];  // layernormed x panel
  __shared__ float sInv[16];

  const int m0 = blockIdx.x * 16;
  const int tid = threadIdx.x;

  if (tid < 16) sInv[tid] = 1.0f / fmaxf(cnt[m0 + tid], 1.0f);
  __syncthreads();

  // stage both 16x512 panels; 2048 float4 slots each, 256 threads -> 8 iters
#pragma unroll
  for (int it = 0; it < 8; ++it) {
    int slot = tid + 256 * it;
    int r = slot >> 7;           // row 0..15
    int c4 = slot & 127;         // float4 col
    float inv = sInv[r];
    float4 a = *(const float4*)(summed + (size_t)(m0 + r) * D + c4 * 4);
    a.x *= inv; a.y *= inv; a.z *= inv; a.w *= inv;
    *(float4*)(&sA[r * LDS_STRIDE + c4 * 4]) = a;
    float4 xv = *(const float4*)(xln + (size_t)(m0 + r) * D + c4 * 4);
    *(float4*)(&sX[r * LDS_STRIDE + c4 * 4]) = xv;
  }
  __syncthreads();

  const int lane  = tid & 31;
  const int waveId = tid >> 5;
  const int lrow  = lane & 15;        // A: M row / B: N col held by this lane
  const int khalf = (lane >> 4) * 2;  // K sub-offset 0 or 2

  for (int j = 0; j < 4; ++j) {
    const int n0 = (waveId * 4 + j) * 16;
    v8f accL = {};
    v8f accR = {};
    const float* wlp = wl + (size_t)(n0 + lrow) * D + khalf;  // B[k][n] = w[n*D+k]
    const float* wrp = wr + (size_t)(n0 + lrow) * D + khalf;
    const float* aP = sA + lrow * LDS_STRIDE + khalf;
    const float* xP = sX + lrow * LDS_STRIDE + khalf;
#pragma unroll 4
    for (int k = 0; k < D; k += 4) {
      v2f aA = *(const v2f*)(aP + k);
      v2f aX = *(const v2f*)(xP + k);
      v2f bL = *(const v2f*)(wlp + k);
      v2f bR = *(const v2f*)(wrp + k);
      accL = __builtin_amdgcn_wmma_f32_16x16x4_f32(false, aA, false, bL,
                                                   (short)0, accL, false, false);
      accR = __builtin_amdgcn_wmma_f32_16x16x4_f32(false, aX, false, bR,
                                                   (short)0, accR, false, false);
    }
    const int col = n0 + lrow;
    const float bias = bl[col];
#pragma unroll
    for (int r = 0; r < 8; ++r) {
      int row = m0 + r + 8 * (lane >> 4);
      float v = fmaxf(accL[r] + accR[r] + bias, 0.0f);
      out[(size_t)row * D + col] = v + resid[(size_t)row * D + col];
    }
  }
}

extern "C" void kernel_launch(void* const* d_in, const int* in_sizes, int n_in,
                              void* d_out, int out_size, void* d_ws, size_t ws_size,
                              hipStream_t stream) {
  const float* nodef = (const float*)d_in[0];
  const int*   ei    = (const int*)d_in[1];
  const float* gamma = (const float*)d_in[2];
  const float* beta  = (const float*)d_in[3];
  const float* wl    = (const float*)d_in[4];
  const float* bl    = (const float*)d_in[5];
  const float* wr    = (const float*)d_in[6];
  float* out = (float*)d_out;

  const int N = in_sizes[0] / D;       // 10000
  const long long E = in_sizes[1] / 2; // 150000

  float* x      = (float*)d_ws;                 // N*D
  float* summed = x + (size_t)N * D;            // N*D
  float* cntf   = summed + (size_t)N * D;       // N

  const int zn = N * D + N;
  gnn_zero_kernel<<<(zn + 255) / 256, 256, 0, stream>>>(summed, zn);
  gnn_ln_kernel<<<(N + 7) / 8, 256, 0, stream>>>(nodef, gamma, beta, x, N);
  const long long sthreads = E * 128;
  gnn_scatter_kernel<<<(int)((sthreads + 255) / 256), 256, 0, stream>>>(x, ei, summed, cntf, E);
  gnn_gemm_kernel<<<N / 16, 256, 0, stream>>>(x, summed, cntf, wl, wr, bl, nodef, out);
}